// ContKDLoss_24773371363804
// MI455X (gfx1250) — compile-verified
//
#include <hip/hip_runtime.h>
#include <math.h>

#define BATCH 2048
#define DIM   4096
#define NTOT  (2 * BATCH)     // 4096 rows of concat(target, surrogate)
#define INV_T 2.0f            // 1 / TEMPERATURE (T = 0.5)

typedef float v2f __attribute__((ext_vector_type(2)));
typedef float v8f __attribute__((ext_vector_type(8)));
typedef int   v4i __attribute__((ext_vector_type(4)));

// Async global->LDS DMA availability (gfx1250); fallback = register staging.
#if defined(__has_builtin)
#if __has_builtin(__builtin_amdgcn_global_load_async_to_lds_b128)
#define HAVE_ASYNC_LDS 1
#endif
#endif
#ifndef HAVE_ASYNC_LDS
#define HAVE_ASYNC_LDS 0
#endif

#if HAVE_ASYNC_LDS
// 16-byte async DMA: global (AS1) -> LDS (AS3), tracked by ASYNCcnt.
__device__ __forceinline__ void asyncCopy16(const float* g, float* l) {
  __builtin_amdgcn_global_load_async_to_lds_b128(
      (__attribute__((address_space(1))) v4i*)g,
      (__attribute__((address_space(3))) v4i*)l, 0, 0);
}
#endif

// ---------------- reduction helpers (wave32) ----------------
__device__ __forceinline__ float waveSum(float v) {
#pragma unroll
  for (int m = 1; m < 32; m <<= 1) v += __shfl_xor(v, m, 32);
  return v;
}
__device__ __forceinline__ float waveMax(float v) {
#pragma unroll
  for (int m = 1; m < 32; m <<= 1) v = fmaxf(v, __shfl_xor(v, m, 32));
  return v;
}
__device__ float blockSum(float v, float* s) {   // 256 threads = 8 waves
  v = waveSum(v);
  __syncthreads();
  if ((threadIdx.x & 31) == 0) s[threadIdx.x >> 5] = v;
  __syncthreads();
  float r = s[0];
#pragma unroll
  for (int w = 1; w < 8; ++w) r += s[w];
  return r;
}
__device__ float blockMax(float v, float* s) {
  v = waveMax(v);
  __syncthreads();
  if ((threadIdx.x & 31) == 0) s[threadIdx.x >> 5] = v;
  __syncthreads();
  float r = s[0];
#pragma unroll
  for (int w = 1; w < 8; ++w) r = fmaxf(r, s[w]);
  return r;
}

// ---------------- Kernel 1: per-row norms, KD stats, MSE ----------------
__global__ __launch_bounds__(256)
void row_stats_kernel(const float* __restrict__ tgt, const float* __restrict__ sur,
                      float* __restrict__ rnorm, float* __restrict__ scal) {
  __shared__ float sh_t[DIM];
  __shared__ float sh_s[DIM];
  __shared__ float red[8];
  const int r   = blockIdx.x;
  const int tid = threadIdx.x;
  const float4* t4p = (const float4*)(tgt + (size_t)r * DIM);
  const float4* s4p = (const float4*)(sur + (size_t)r * DIM);

  float ss_t = 0.f, ss_s = 0.f, sd2 = 0.f;
  float mt = -INFINITY, ms = -INFINITY;
  for (int c = tid; c < DIM / 4; c += 256) {
    float4 t4 = t4p[c], s4 = s4p[c];
    ((float4*)sh_t)[c] = t4;
    ((float4*)sh_s)[c] = s4;
    ss_t += t4.x * t4.x + t4.y * t4.y + t4.z * t4.z + t4.w * t4.w;
    ss_s += s4.x * s4.x + s4.y * s4.y + s4.z * s4.z + s4.w * s4.w;
    float dx = s4.x - t4.x, dy = s4.y - t4.y, dz = s4.z - t4.z, dw = s4.w - t4.w;
    sd2 += dx * dx + dy * dy + dz * dz + dw * dw;
    mt = fmaxf(mt, fmaxf(fmaxf(t4.x, t4.y), fmaxf(t4.z, t4.w)));
    ms = fmaxf(ms, fmaxf(fmaxf(s4.x, s4.y), fmaxf(s4.z, s4.w)));
  }
  __syncthreads();  // LDS rows complete
  ss_t = blockSum(ss_t, red);
  ss_s = blockSum(ss_s, red);
  sd2  = blockSum(sd2, red);
  mt   = blockMax(mt, red);
  ms   = blockMax(ms, red);

  float et_sum = 0.f, es_sum = 0.f, wsum = 0.f;
  for (int c = tid; c < DIM; c += 256) {
    float t = sh_t[c], s = sh_s[c];
    float et = __expf(t - mt);
    et_sum += et;
    es_sum += __expf(s - ms);
    wsum   += et * (t - s);
  }
  et_sum = blockSum(et_sum, red);
  es_sum = blockSum(es_sum, red);
  wsum   = blockSum(wsum, red);

  if (tid == 0) {
    float nt = fmaxf(sqrtf(ss_t), 1e-8f);
    float ns = fmaxf(sqrtf(ss_s), 1e-8f);
    rnorm[r]         = 1.f / nt;
    rnorm[BATCH + r] = 1.f / ns;
    float lse_t = mt + __logf(et_sum);
    float lse_s = ms + __logf(es_sum);
    float kd_row = wsum / et_sum - lse_t + lse_s;   // KD_T = 1
    atomicAdd(&scal[0], kd_row);
    atomicAdd(&scal[1], sd2);
  }
}

// ---------------- Kernel 2: fused symmetric similarity GEMM (f32 WMMA) ------
#define BM 128
#define BN 128
#define NB (NTOT / BM)              // 32 block-rows
#define NBLK (NB * (NB + 1) / 2)    // 528 triangular blocks

#if HAVE_ASYNC_LDS
#define BK  16
#define LDT 20   // 80B rows: float4/float2 aligned, padded vs bank conflicts
#define GEMM_BOUNDS __launch_bounds__(256)
#else
#define BK  32
#define LDT 36   // 144B rows
#define GEMM_BOUNDS __launch_bounds__(256, 1)  // full VGPR budget: no staging spills
#endif
#define NST (DIM / BK)

__device__ __forceinline__ void computeStage(const float* As, const float* Bs,
                                             int wvM, int wvN, int lr, int lhi,
                                             v8f (&acc)[4][2]) {
#pragma unroll
  for (int k4 = 0; k4 < BK; k4 += 4) {
    v2f afr[4], bfr[2];
#pragma unroll
    for (int mi = 0; mi < 4; ++mi)
      afr[mi] = *(const v2f*)(&As[(wvM + mi * 16 + lr) * LDT + k4 + 2 * lhi]);
#pragma unroll
    for (int nj = 0; nj < 2; ++nj)
      bfr[nj] = *(const v2f*)(&Bs[(wvN + nj * 16 + lr) * LDT + k4 + 2 * lhi]);
#pragma unroll
    for (int mi = 0; mi < 4; ++mi)
#pragma unroll
      for (int nj = 0; nj < 2; ++nj)
        acc[mi][nj] = __builtin_amdgcn_wmma_f32_16x16x4_f32(
            false, afr[mi], false, bfr[nj], (short)0, acc[mi][nj], false, false);
  }
}

__global__ GEMM_BOUNDS
void sim_gemm_kernel(const float* __restrict__ tgt, const float* __restrict__ sur,
                     const float* __restrict__ rnorm,
                     float* __restrict__ rowsum, float* __restrict__ pos) {
  __shared__ float rs[BM];
  __shared__ float cs[BN];

  // --- decode triangular block id: f(b) = NB*b - b*(b-1)/2 blocks before row b
  const int L = blockIdx.x;
  int bi = 0;
  while ((bi + 1) * NB - ((bi + 1) * bi) / 2 <= L) ++bi;
  const int bj = bi + (L - (bi * NB - (bi * (bi - 1)) / 2));
  const int rowBase = bi * BM;
  const int colBase = bj * BN;
  const bool diag = (bi == bj);

  const int tid = threadIdx.x;
  if (tid < BM) { rs[tid] = 0.f; cs[tid] = 0.f; }

  const int wave = tid >> 5;
  const int lane = tid & 31;
  const int lhi  = lane >> 4;
  const int lr   = lane & 15;
  const int wvM  = (wave & 1) * 64;    // 2 waves along M
  const int wvN  = (wave >> 1) * 32;   // 4 waves along N

  v8f acc[4][2] = {};                  // 64x32 wave tile

#if HAVE_ASYNC_LDS
  // ===== async DMA path: global -> LDS with no VGPR staging, double buffer ==
  __shared__ float As[2 * BM * LDT];
  __shared__ float Bs[2 * BM * LDT];
  const int ldRow = tid >> 2;          // 0..63
  const int ldCol = (tid & 3) * 4;     // float4 col inside 16-wide k slice

  auto issueStage = [&](int st) {      // 4 async b128 ops per thread-stage
    const int kk = st * BK;
    const int bo = (st & 1) * (BM * LDT);
#pragma unroll
    for (int it = 0; it < 2; ++it) {
      int rA = ldRow + it * 64;
      int gi = rowBase + rA;
      const float* srcA = (gi < BATCH) ? (tgt + (size_t)gi * DIM)
                                       : (sur + (size_t)(gi - BATCH) * DIM);
      asyncCopy16(srcA + kk + ldCol, &As[bo + rA * LDT + ldCol]);
      int gj = colBase + rA;
      const float* srcB = (gj < BATCH) ? (tgt + (size_t)gj * DIM)
                                       : (sur + (size_t)(gj - BATCH) * DIM);
      asyncCopy16(srcB + kk + ldCol, &Bs[bo + rA * LDT + ldCol]);
    }
  };

  issueStage(0);
  issueStage(1);
  for (int st = 0; st < NST; ++st) {
    if (st + 1 < NST)
      asm volatile("s_wait_asynccnt 0x4" ::: "memory");  // stage st landed
    else
      asm volatile("s_wait_asynccnt 0x0" ::: "memory");
    __syncthreads();                                     // publish buffer st&1
    const int bo = (st & 1) * (BM * LDT);
    computeStage(&As[bo], &Bs[bo], wvM, wvN, lr, lhi, acc);
    if (st + 2 < NST) {
      __syncthreads();                                   // all reads of buf done
      issueStage(st + 2);                                // safe to overwrite
    }
  }
#else
  // ===== fallback: register-staged prefetch, single LDS buffer ==============
  __shared__ float As[BM * LDT];
  __shared__ float Bs[BM * LDT];
  const int ldRow = tid >> 3;          // 0..31
  const int ldCol = (tid & 7) * 4;     // float4 col inside 32-wide k slice

  float4 ra[4], rb[4];
  auto loadStage = [&](int kk) {
#pragma unroll
    for (int it = 0; it < 4; ++it) {
      int rA = ldRow + it * 32;
      int gi = rowBase + rA;
      const float* srcA = (gi < BATCH) ? (tgt + (size_t)gi * DIM)
                                       : (sur + (size_t)(gi - BATCH) * DIM);
      ra[it] = *(const float4*)(srcA + kk + ldCol);
      int gj = colBase + rA;
      const float* srcB = (gj < BATCH) ? (tgt + (size_t)gj * DIM)
                                       : (sur + (size_t)(gj - BATCH) * DIM);
      rb[it] = *(const float4*)(srcB + kk + ldCol);
    }
  };
  auto storeStage = [&]() {
#pragma unroll
    for (int it = 0; it < 4; ++it) {
      int rA = ldRow + it * 32;
      *(float4*)(&As[rA * LDT + ldCol]) = ra[it];
      *(float4*)(&Bs[rA * LDT + ldCol]) = rb[it];
    }
  };

  loadStage(0);
  for (int st = 0; st < NST; ++st) {
    storeStage();
    if (st + 1 < NST) loadStage((st + 1) * BK);
    __syncthreads();
    computeStage(As, Bs, wvM, wvN, lr, lhi, acc);
    __syncthreads();
  }
#endif

  // ---- epilogue: scale by rnorm, exp row/col sums, positives ----
  float rnj[2];
  rnj[0] = rnorm[colBase + wvN + lr];
  rnj[1] = rnorm[colBase + wvN + 16 + lr];
  float ecol[2] = {0.f, 0.f};
#pragma unroll
  for (int mi = 0; mi < 4; ++mi) {
    int ibase = rowBase + wvM + mi * 16 + 8 * lhi;  // C layout: VGPR v -> M=v (+8 hi half)
#pragma unroll
    for (int v = 0; v < 8; ++v) {
      int i = ibase + v;
      float rni = rnorm[i];
      float p = 0.f;
#pragma unroll
      for (int nj = 0; nj < 2; ++nj) {
        int j = colBase + wvN + nj * 16 + lr;
        float simv = acc[mi][nj][v] * rni * rnj[nj];
        if (i < BATCH && j == i + BATCH) {          // positive pair (i, i+B)
          pos[i] = simv;
          pos[j] = simv;
        }
        float e = (i == j) ? 0.f : __expf(simv * INV_T);
        p += e;
        ecol[nj] += e;
      }
#pragma unroll
      for (int m = 1; m < 16; m <<= 1) p += __shfl_xor(p, m, 32);
      if (lr == 0) atomicAdd(&rs[wvM + mi * 16 + 8 * lhi + v], p);
    }
  }
  if (!diag) {  // symmetric contribution: column sums feed rowsum of bj rows
#pragma unroll
    for (int nj = 0; nj < 2; ++nj) {
      float c = ecol[nj] + __shfl_xor(ecol[nj], 16, 32);
      if (lhi == 0) atomicAdd(&cs[wvN + nj * 16 + lr], c);
    }
  }
  __syncthreads();
  if (tid < BM) {
    atomicAdd(&rowsum[rowBase + tid], rs[tid]);
    if (!diag) atomicAdd(&rowsum[colBase + tid], cs[tid]);
  }
}

// ---------------- Kernel 3: finalize ----------------
__global__ __launch_bounds__(256)
void finalize_kernel(const float* __restrict__ rowsum, const float* __restrict__ pos,
                     const float* __restrict__ scal, float* __restrict__ out) {
  __shared__ float red[8];
  float acc = 0.f;
  for (int i = threadIdx.x; i < NTOT; i += 256)
    acc += __logf(rowsum[i]) - pos[i] * INV_T;   // log(denom) - pos/T
  acc = blockSum(acc, red);
  if (threadIdx.x == 0) {
    float cont = acc / (float)NTOT;
    float kd   = scal[0] / (float)BATCH;                 // KD_T^2 = 1
    float mse  = scal[1] / ((float)BATCH * (float)DIM);
    out[0] = cont + kd + mse;
  }
}

// ---------------- launch ----------------
extern "C" void kernel_launch(void* const* d_in, const int* in_sizes, int n_in,
                              void* d_out, int out_size, void* d_ws, size_t ws_size,
                              hipStream_t stream) {
  const float* tgt = (const float*)d_in[0];
  const float* sur = (const float*)d_in[1];
  float* ws     = (float*)d_ws;
  float* rnorm  = ws;                 // NTOT floats
  float* rowsum = ws + NTOT;          // NTOT floats (accumulated)
  float* pos    = ws + 2 * NTOT;      // NTOT floats
  float* scal   = ws + 3 * NTOT;      // 2 floats: kd_sum, mse_sum

  (void)hipMemsetAsync(rowsum, 0, (size_t)(2 * NTOT + 2) * sizeof(float), stream);

  row_stats_kernel<<<BATCH, 256, 0, stream>>>(tgt, sur, rnorm, scal);
  sim_gemm_kernel<<<NBLK, 256, 0, stream>>>(tgt, sur, rnorm, rowsum, pos);
  finalize_kernel<<<1, 256, 0, stream>>>(rowsum, pos, scal, (float*)d_out);
}